// ClusterNet_BGC_Contrast_75642964017170
// MI455X (gfx1250) — compile-verified
//
#include <hip/hip_runtime.h>
#include <hip/hip_bf16.h>
#include <math.h>

typedef __attribute__((ext_vector_type(2))) float v2f;
typedef __attribute__((ext_vector_type(8))) float v8f;

#define BD 8
#define DD 64
#define HH 192
#define NN 36864          // 192*192
#define KK 32
#define PB 144            // pixel blocks of 256 per batch
#define GB 288            // gemm blocks of 128 rows per batch
#define INV_EPS 20.0f     // 1/0.05
#define NUMPIX 2268.0f    // boundary pixel count

// ---------------- transpose z (b,d,n) -> zfn (b,n,d) with per-pixel L2 norm ----
__global__ __launch_bounds__(256) void transpose_norm_kernel(
    const float* __restrict__ z, float* __restrict__ zfn, float* __restrict__ norms) {
  __shared__ float tile[64 * 65];
  __shared__ float rn[64];
  const int b = blockIdx.y, t = threadIdx.x;
  const int pbase = blockIdx.x * 64;
  const float* zb = z + (size_t)b * DD * NN;
#pragma unroll
  for (int i = 0; i < 16; ++i) {
    int idx = i * 256 + t;
    int c = idx >> 6, pp = idx & 63;
    tile[c * 65 + pp] = zb[(size_t)c * NN + pbase + pp];
  }
  __syncthreads();
  if (t < 64) {
    float ss = 0.f;
#pragma unroll
    for (int c = 0; c < 64; ++c) { float x = tile[c * 65 + t]; ss += x * x; }
    float nr = sqrtf(ss);
    norms[(size_t)b * NN + pbase + t] = nr;
    rn[t] = 1.0f / fmaxf(nr, 1e-12f);
  }
  __syncthreads();
#pragma unroll
  for (int i = 0; i < 16; ++i) {
    int idx = i * 256 + t;
    int pp = idx >> 6, c = idx & 63;
    zfn[((size_t)b * NN + pbase + pp) * 64 + c] = tile[c * 65 + pp] * rn[pp];
  }
}

// ---------------- boundary background mean per (b,c) ----------------
__global__ __launch_bounds__(256) void bg_kernel(const float* __restrict__ z,
                                                 float* __restrict__ bg) {
  __shared__ float red[256];
  const int c = blockIdx.x, b = blockIdx.y, t = threadIdx.x;
  const float* zp = z + ((size_t)b * DD + c) * NN;
  float a = 0.f;
  for (int p = t; p < NN; p += 256) {
    int r = p / HH, col = p - r * HH;
    if (r < 3 || r >= HH - 3 || col < 3 || col >= HH - 3) a += zp[p];
  }
  red[t] = a; __syncthreads();
  for (int s = 128; s > 0; s >>= 1) { if (t < s) red[t] += red[t + s]; __syncthreads(); }
  if (t == 0) bg[b * 64 + c] = red[0] / NUMPIX;
}

// ---------------- row L2-normalize 64-wide vectors (bg_hat / protos) ----------
__global__ __launch_bounds__(64) void rownorm64_kernel(const float* __restrict__ src,
                                                       float* __restrict__ dst) {
  __shared__ float red[64];
  const int row = blockIdx.x, t = threadIdx.x;
  float x = src[(size_t)row * 64 + t];
  red[t] = x * x; __syncthreads();
  for (int s = 32; s > 0; s >>= 1) { if (t < s) red[t] += red[t + s]; __syncthreads(); }
  float inv = 1.0f / fmaxf(sqrtf(red[0]), 1e-12f);
  dst[(size_t)row * 64 + t] = x * inv;
}

// ---------------- saliency similarity + block min/max partials ----------------
__global__ __launch_bounds__(256) void sim_kernel(const float* __restrict__ zfn,
    const float* __restrict__ bgh, float* __restrict__ sim,
    float* __restrict__ pmin, float* __restrict__ pmax) {
  __shared__ float g[64];
  __shared__ float rmin[256], rmax[256];
  const int b = blockIdx.y, t = threadIdx.x;
  if (t < 64) g[t] = bgh[b * 64 + t];
  __syncthreads();
  const int p = blockIdx.x * 256 + t;
  const float* zr = zfn + ((size_t)b * NN + p) * 64;
  float d = 0.f;
#pragma unroll
  for (int c = 0; c < 64; ++c) d += zr[c] * g[c];
  sim[(size_t)b * NN + p] = d;
  rmin[t] = d; rmax[t] = d; __syncthreads();
  for (int s = 128; s > 0; s >>= 1) {
    if (t < s) { rmin[t] = fminf(rmin[t], rmin[t + s]); rmax[t] = fmaxf(rmax[t], rmax[t + s]); }
    __syncthreads();
  }
  if (t == 0) { pmin[b * PB + blockIdx.x] = rmin[0]; pmax[b * PB + blockIdx.x] = rmax[0]; }
}

__global__ __launch_bounds__(256) void minmax2_kernel(const float* __restrict__ pmin,
    const float* __restrict__ pmax, float* __restrict__ sstat) {
  __shared__ float rmin[256], rmax[256];
  const int b = blockIdx.x, t = threadIdx.x;
  float mn = 3.4e38f, mx = -3.4e38f;
  for (int i = t; i < PB; i += 256) { mn = fminf(mn, pmin[b * PB + i]); mx = fmaxf(mx, pmax[b * PB + i]); }
  rmin[t] = mn; rmax[t] = mx; __syncthreads();
  for (int s = 128; s > 0; s >>= 1) {
    if (t < s) { rmin[t] = fminf(rmin[t], rmin[t + s]); rmax[t] = fmaxf(rmax[t], rmax[t + s]); }
    __syncthreads();
  }
  if (t == 0) { sstat[b * 2] = rmin[0]; sstat[b * 2 + 1] = rmax[0]; }
}

__global__ __launch_bounds__(256) void mask_kernel(const float* __restrict__ sim,
    const float* __restrict__ sstat, float* __restrict__ out_mask) {
  const int b = blockIdx.y;
  const int p = blockIdx.x * 256 + threadIdx.x;
  float smin = sstat[b * 2], smax = sstat[b * 2 + 1];
  float s = sim[(size_t)b * NN + p];
  out_mask[(size_t)b * NN + p] = 1.0f - (s - smin) / (smax - smin);
}

// ------- WMMA GEMM: s = zfn(n x 64) @ protos^T(64 x 32), fused softmax+exp ----
__global__ __launch_bounds__(128) void gemm_softmax_exp_kernel(
    const float* __restrict__ zfn, const float* __restrict__ protos,
    float* __restrict__ E) {
  __shared__ float smem[128 * 33];
  const int b = blockIdx.y, t = threadIdx.x;
  const int wave = t >> 5, lane = t & 31;
  const int l15 = lane & 15, lhi = lane >> 4;
  const int rowBase = blockIdx.x * 128 + wave * 32;
  const float* zb = zfn + (size_t)b * NN * 64;
  const float* pb = protos + (size_t)b * (KK * 64);

  // A frag (16x4 f32): lane<16 holds K=0,1; lane>=16 holds K=2,3 of row (base+l15)
  const float* A0 = zb + (size_t)(rowBase + l15) * 64 + lhi * 2;
  const float* A1 = A0 + 16 * 64;
  // B frag (4x16 f32): B[kin][nn] = protos[nn][kc+kin]; same striping pattern
  const float* B0 = pb + l15 * 64 + lhi * 2;
  const float* B1 = B0 + 16 * 64;

  v8f acc00 = {}; v8f acc01 = {}; v8f acc10 = {}; v8f acc11 = {};
#pragma unroll
  for (int kc = 0; kc < 64; kc += 4) {
    v2f a0 = *(const v2f*)(A0 + kc);
    v2f a1 = *(const v2f*)(A1 + kc);
    v2f b0 = *(const v2f*)(B0 + kc);
    v2f b1 = *(const v2f*)(B1 + kc);
    acc00 = __builtin_amdgcn_wmma_f32_16x16x4_f32(false, a0, false, b0, (short)0, acc00, false, false);
    acc01 = __builtin_amdgcn_wmma_f32_16x16x4_f32(false, a0, false, b1, (short)0, acc01, false, false);
    acc10 = __builtin_amdgcn_wmma_f32_16x16x4_f32(false, a1, false, b0, (short)0, acc10, false, false);
    acc11 = __builtin_amdgcn_wmma_f32_16x16x4_f32(false, a1, false, b1, (short)0, acc11, false, false);
  }
  // C/D layout: VGPR j -> rows (j, j+8) split across lane halves, col = l15
  const int r0 = wave * 32 + lhi * 8;
#pragma unroll
  for (int j = 0; j < 8; ++j) {
    smem[(r0 + j) * 33 + l15]           = acc00[j];
    smem[(r0 + j) * 33 + 16 + l15]      = acc01[j];
    smem[(r0 + 16 + j) * 33 + l15]      = acc10[j];
    smem[(r0 + 16 + j) * 33 + 16 + l15] = acc11[j];
  }
  __syncthreads();
  // per-row softmax over K=32, then E = exp(softmax / eps)
  float vals[32];
  float m = -3.4e38f;
#pragma unroll
  for (int k = 0; k < 32; ++k) { vals[k] = smem[t * 33 + k]; m = fmaxf(m, vals[k]); }
  float ssum = 0.f;
#pragma unroll
  for (int k = 0; k < 32; ++k) { vals[k] = expf(vals[k] - m); ssum += vals[k]; }
  float inv = 1.0f / ssum;
  float* Erow = E + ((size_t)b * NN + blockIdx.x * 128 + t) * 32;
#pragma unroll
  for (int k = 0; k < 32; ++k) Erow[k] = expf(vals[k] * inv * INV_EPS);
}

// ---------------- sinkhorn: row-sum partials R_k = sum_p E*v ------------------
__global__ __launch_bounds__(256) void sink_rowpart_kernel(const float* __restrict__ E,
    const float* __restrict__ v, float* __restrict__ Rpart, int use_v) {
  __shared__ float red[8][32];
  const int b = blockIdx.y, t = threadIdx.x;
  const int k = t & 31, g = t >> 5;
  const int pbase = blockIdx.x * 256 + g * 32;
  float a = 0.f;
  for (int j = 0; j < 32; ++j) {
    int p = pbase + j;
    float vp = use_v ? v[(size_t)b * NN + p] : 1.0f;
    a += E[((size_t)b * NN + p) * 32 + k] * vp;
  }
  red[g][k] = a; __syncthreads();
  if (t < 32) {
    float s = 0.f;
#pragma unroll
    for (int gg = 0; gg < 8; ++gg) s += red[gg][t];   // fixed order: deterministic
    Rpart[((size_t)b * PB + blockIdx.x) * 32 + t] = s;
  }
}

__global__ __launch_bounds__(32) void sink_u_kernel(const float* __restrict__ Rpart,
                                                    float* __restrict__ u) {
  const int b = blockIdx.x, k = threadIdx.x;
  float s = 0.f;
  for (int blk = 0; blk < PB; ++blk) s += Rpart[((size_t)b * PB + blk) * 32 + k];
  u[b * 32 + k] = 1.0f / (32.0f * s);
}

__global__ __launch_bounds__(256) void sink_v_kernel(const float* __restrict__ E,
    const float* __restrict__ u, float* __restrict__ v) {
  __shared__ float us[32];
  const int b = blockIdx.y, t = threadIdx.x;
  if (t < 32) us[t] = u[b * 32 + t];
  __syncthreads();
  const int p = blockIdx.x * 256 + t;
  const float* Erow = E + ((size_t)b * NN + p) * 32;
  float s = 0.f;
#pragma unroll
  for (int k = 0; k < 32; ++k) s += Erow[k] * us[k];
  v[(size_t)b * NN + p] = 1.0f / ((float)NN * s);
}

// -------- finalize: Q = E*u*v*N, logits/indexes out, proto partial sums -------
__global__ __launch_bounds__(256) void finalize_kernel(const float* __restrict__ E,
    const float* __restrict__ u, const float* __restrict__ v,
    const float* __restrict__ zfn, const float* __restrict__ norms,
    float* __restrict__ out_logits, float* __restrict__ out_idx,
    float* __restrict__ nppart, float* __restrict__ cpart) {
  __shared__ float us[32];
  __shared__ float acc[32 * 64];
  __shared__ float cnt[32];
  const int b = blockIdx.y, t = threadIdx.x, blk = blockIdx.x;
  if (t < 32) { us[t] = u[b * 32 + t]; cnt[t] = 0.f; }
#pragma unroll
  for (int i = 0; i < 8; ++i) acc[i * 256 + t] = 0.f;
  __syncthreads();
  const int p = blk * 256 + t;
  const float* Erow = E + ((size_t)b * NN + p) * 32;
  const float vpN = v[(size_t)b * NN + p] * (float)NN;
  float best = -1.f; int bi = 0;
#pragma unroll
  for (int k = 0; k < 32; ++k) {
    float q = Erow[k] * us[k] * vpN;
    out_logits[((size_t)b * KK + k) * NN + p] = q;
    if (q > best) { best = q; bi = k; }
  }
  out_idx[(size_t)b * NN + p] = (float)bi;
  const float nrm = norms[(size_t)b * NN + p];
  const float* zr = zfn + ((size_t)b * NN + p) * 64;
#pragma unroll
  for (int c = 0; c < 64; ++c) atomicAdd(&acc[bi * 64 + c], zr[c] * nrm);
  atomicAdd(&cnt[bi], 1.0f);
  __syncthreads();
  float* np = nppart + (((size_t)b * PB + blk) * 32) * 64;
#pragma unroll
  for (int i = 0; i < 8; ++i) np[i * 256 + t] = acc[i * 256 + t];
  if (t < 32) cpart[((size_t)b * PB + blk) * 32 + t] = cnt[t];
}

// -------- reduce proto partials, mean, nan_to_num, L2 normalize ---------------
__global__ __launch_bounds__(64) void proto_update_kernel(const float* __restrict__ nppart,
    const float* __restrict__ cpart, float* __restrict__ protos_cur,
    float* __restrict__ out_protos) {
  __shared__ float red[64];
  const int row = blockIdx.x;            // b*32 + k
  const int b = row >> 5, k = row & 31, t = threadIdx.x;
  float s = 0.f, cc = 0.f;
  for (int blk = 0; blk < PB; ++blk) {   // fixed order: deterministic
    s  += nppart[(((size_t)b * PB + blk) * 32 + k) * 64 + t];
    cc += cpart[((size_t)b * PB + blk) * 32 + k];
  }
  float mean = (cc > 0.f) ? (s / cc) : 0.f;   // nan_to_num(0/0)=0
  red[t] = mean * mean; __syncthreads();
  for (int st = 32; st > 0; st >>= 1) { if (t < st) red[t] += red[t + st]; __syncthreads(); }
  float inv = 1.0f / fmaxf(sqrtf(red[0]), 1e-12f);
  float val = mean * inv;
  protos_cur[(size_t)row * 64 + t] = val;
  out_protos[(size_t)row * 64 + t] = val;
}

// =============================== host launch ==================================
extern "C" void kernel_launch(void* const* d_in, const int* in_sizes, int n_in,
                              void* d_out, int out_size, void* d_ws, size_t ws_size,
                              hipStream_t stream) {
  (void)in_sizes; (void)n_in; (void)out_size; (void)ws_size;
  const float* z  = (const float*)d_in[0];
  const float* p0 = (const float*)d_in[1];
  float* out = (float*)d_out;

  // output offsets (floats)
  const size_t O_Z = 0;
  const size_t O_MASK = (size_t)BD * DD * NN;              // 18,874,368
  const size_t O_LOGITS = O_MASK + (size_t)BD * NN;        // + 294,912
  const size_t O_IDX = O_LOGITS + (size_t)BD * KK * NN;    // + 9,437,184
  const size_t O_PROTOS = O_IDX + (size_t)BD * NN;         // + 294,912

  // workspace layout (floats)
  float* w = (float*)d_ws;
  size_t o = 0;
  float* zfn   = w + o; o += (size_t)BD * NN * 64;
  float* norms = w + o; o += (size_t)BD * NN;
  float* E     = w + o; o += (size_t)BD * NN * 32;
  float* v     = w + o; o += (size_t)BD * NN;
  float* sim   = w + o; o += (size_t)BD * NN;
  float* bg    = w + o; o += (size_t)BD * 64;
  float* bgh   = w + o; o += (size_t)BD * 64;
  float* pcur  = w + o; o += (size_t)BD * KK * 64;
  float* u     = w + o; o += (size_t)BD * KK;
  float* Rpart = w + o; o += (size_t)BD * PB * KK;
  float* pmin  = w + o; o += (size_t)BD * PB;
  float* pmax  = w + o; o += (size_t)BD * PB;
  float* sstat = w + o; o += (size_t)BD * 2;
  float* nppart= w + o; o += (size_t)BD * PB * KK * 64;
  float* cpart = w + o; o += (size_t)BD * PB * KK;

  // output[0] = z passthrough
  hipMemcpyAsync(out + O_Z, z, (size_t)BD * DD * NN * sizeof(float),
                 hipMemcpyDeviceToDevice, stream);

  // features: transpose + per-pixel L2 norm
  transpose_norm_kernel<<<dim3(NN / 64, BD), 256, 0, stream>>>(z, zfn, norms);

  // saliency
  bg_kernel<<<dim3(64, BD), 256, 0, stream>>>(z, bg);
  rownorm64_kernel<<<BD, 64, 0, stream>>>(bg, bgh);
  sim_kernel<<<dim3(PB, BD), 256, 0, stream>>>(zfn, bgh, sim, pmin, pmax);
  minmax2_kernel<<<BD, 256, 0, stream>>>(pmin, pmax, sstat);
  mask_kernel<<<dim3(PB, BD), 256, 0, stream>>>(sim, sstat, out + O_MASK);

  // prototypes init normalize
  rownorm64_kernel<<<BD * KK, 64, 0, stream>>>(p0, pcur);

  // cluster iterations
  for (int it = 0; it < 3; ++it) {
    gemm_softmax_exp_kernel<<<dim3(GB, BD), 128, 0, stream>>>(zfn, pcur, E);
    for (int t = 0; t < 3; ++t) {
      sink_rowpart_kernel<<<dim3(PB, BD), 256, 0, stream>>>(E, v, Rpart, t > 0 ? 1 : 0);
      sink_u_kernel<<<BD, 32, 0, stream>>>(Rpart, u);
      sink_v_kernel<<<dim3(PB, BD), 256, 0, stream>>>(E, u, v);
    }
    finalize_kernel<<<dim3(PB, BD), 256, 0, stream>>>(
        E, u, v, zfn, norms, out + O_LOGITS, out + O_IDX, nppart, cpart);
    proto_update_kernel<<<BD * KK, 64, 0, stream>>>(nppart, cpart, pcur, out + O_PROTOS);
  }
}